// PointNetLoss_66915590471777
// MI455X (gfx1250) — compile-verified
//
#include <hip/hip_runtime.h>
#include <hip/hip_bf16.h>

#define BB 8
#define NN 2048
#define MM 2048
#define JJ 21
#define NW 778

typedef __attribute__((ext_vector_type(2))) float v2f;
typedef __attribute__((ext_vector_type(8))) float v8f;

// JOINT_RANGES split into lo / hi, row-major (21 x 3)
__constant__ float c_lo[JJ * 3] = {
    -1.57f, -1.57f, -1.57f,
    -1.05f, 0.f, -0.79f,
    -0.52f, 0.f, -0.35f,
    -0.26f, 0.f, 0.f,
    -1.05f, 0.f, 0.f,
    0.f, 0.f, 0.f,
    -1.05f, 0.f, -0.52f,
    -0.52f, 0.f, 0.f,
    -0.52f, 0.f, 0.f,
    0.f, 0.f, 0.f,
    -1.05f, 0.f, -0.52f,
    -0.52f, 0.f, 0.f,
    -0.52f, 0.f, 0.f,
    0.f, 0.f, 0.f,
    -1.05f, 0.f, -0.52f,
    -0.52f, 0.f, 0.f,
    -0.52f, 0.f, 0.f,
    0.f, 0.f, 0.f,
    -1.05f, 0.f, -0.52f,
    -0.52f, 0.f, 0.f,
    -0.52f, 0.f, 0.f};
__constant__ float c_hi[JJ * 3] = {
    1.57f, 1.57f, 1.57f,
    1.05f, 0.f, 0.26f,
    0.52f, 0.f, 0.79f,
    0.26f, 0.f, 0.f,
    0.f, 0.f, 0.f,
    0.f, 0.f, 0.f,
    0.26f, 0.f, 0.26f,
    0.f, 0.f, 0.f,
    0.f, 0.f, 0.f,
    0.f, 0.f, 0.f,
    0.26f, 0.f, 0.26f,
    0.f, 0.f, 0.f,
    0.f, 0.f, 0.f,
    0.f, 0.f, 0.f,
    0.26f, 0.f, 0.26f,
    0.f, 0.f, 0.f,
    0.f, 0.f, 0.f,
    0.f, 0.f, 0.f,
    0.26f, 0.f, 0.26f,
    0.f, 0.f, 0.f,
    0.f, 0.f, 0.f};

#define BITS_100 0x42C80000u   // 100.0f == (10.0f)^2  (masked-min sentinel, squared domain)
#define BITS_INF 0x7F800000u   // +inf

// raw v_sqrt_f32 (no IEEE fixup expansion); only used post-reduction
static __device__ __forceinline__ float raw_sqrtf(float x) {
  return __builtin_amdgcn_sqrtf(x);
}

// VALU xor-16 lane exchange (v_permlanex16_b32, identity nibble selects) —
// avoids ds_bpermute (and its DScnt drain) in the hot loop.
static __device__ __forceinline__ float xor16f(float v) {
  unsigned u = __float_as_uint(v);
  unsigned r = __builtin_amdgcn_permlanex16(u, u, 0x76543210u, 0xfedcba98u, true, false);
  return __uint_as_float(r);
}

// ---------------- init workspace ----------------
__global__ void init_kernel(unsigned* dist2b, unsigned* jd2b, float* jsum1, float* s1) {
  int i = blockIdx.x * blockDim.x + threadIdx.x;
  if (i < BB * MM) {
    dist2b[i] = BITS_INF;
    jd2b[i] = BITS_100;
  }
  if (i < JJ) jsum1[i] = 0.f;
  if (i == 0) *s1 = 0.f;
}

// ---------------- seg = argmin(seg_logits, axis=-1) (first occurrence) ----------------
__global__ void seg_kernel(const float* __restrict__ seg_logits, int* __restrict__ seg) {
  int i = blockIdx.x * blockDim.x + threadIdx.x;
  if (i >= BB * MM) return;
  const float* p = seg_logits + (size_t)i * JJ;
  float mv = p[0];
  int mi = 0;
#pragma unroll
  for (int k = 1; k < JJ; ++k) {
    float v = p[k];
    if (v < mv) { mv = v; mi = k; }
  }
  seg[i] = mi;
}

// ---------------- jwbits[b*N+n] = bitmask_j( bone_weights[j, out_idxs[b,n]] > 0.15 ) ----
__global__ void jw_kernel(const int* __restrict__ out_idxs,
                          const float* __restrict__ bone_weights,
                          unsigned* __restrict__ jwbits) {
  int i = blockIdx.x * blockDim.x + threadIdx.x;
  if (i >= BB * NN) return;
  int idx = out_idxs[i];
  unsigned bits = 0;
#pragma unroll
  for (int j = 0; j < JJ; ++j)
    bits |= (bone_weights[j * NW + idx] > 0.15f) ? (1u << j) : 0u;
  jwbits[i] = bits;
}

// ---------------- main fused chamfer + per-joint masked-min kernel ----------------
// grid = B * (N/16) blocks, 128 threads (4 wave32s). Block owns (b, 16 rows of N),
// waves split the 128 column tiles of M. Gram tile via V_WMMA_F32_16X16X4_F32.
// ALL min reductions run on SQUARED distances (sqrt monotonic); sqrt only after
// reduction. Masked updates are branchless: sentinel 100.0 pushes are min-no-ops,
// and negative float bits (sign set) map to huge uints -> ignored by uint-min,
// so the >=0 clamp is deferred to the push/epilogue points.
__global__ __launch_bounds__(128) void chamfer_kernel(
    const float* __restrict__ gen_pc, const float* __restrict__ target_pc,
    const unsigned* __restrict__ jwbits, const int* __restrict__ seg,
    unsigned* __restrict__ dist2b, unsigned* __restrict__ jd2b,
    float* __restrict__ jsum1, float* __restrict__ s1) {
  const int b = blockIdx.x / (NN / 16);
  const int n0 = (blockIdx.x % (NN / 16)) * 16;
  const int wid = threadIdx.x >> 5;
  const int lane = threadIdx.x & 31;
  const int hf = lane >> 4;      // half of wave: 0 -> rows q, 1 -> rows q+8
  const int lcol = lane & 15;    // column within 16-wide tile

  __shared__ unsigned jd1[4][JJ * 16];  // per-wave masked row-min accumulators (d^2 bits)
  __shared__ unsigned rowminb[16];      // chamfer row mins (d^2 bits)
  __shared__ float jsumLds[JJ];

  for (int s = threadIdx.x; s < 4 * JJ * 16; s += blockDim.x)
    (&jd1[0][0])[s] = BITS_100;
  if (threadIdx.x < 16) rowminb[threadIdx.x] = BITS_INF;
  if (threadIdx.x < JJ) jsumLds[threadIdx.x] = 0.f;
  __syncthreads();

  // A fragment: 16x4 f32. Lanes 0-15 hold K=0,1 of row lcol; lanes 16-31 hold K=2,3.
  const float* gp = gen_pc + ((size_t)b * NN + n0 + lcol) * 3;
  float px = gp[0], py = gp[1], pz = gp[2];
  float xxlane = px * px + py * py + pz * pz;   // |x_row|^2 for row lcol
  v2f a;
  a.x = hf ? pz : px;
  a.y = hf ? 0.f : py;

  float xx8[8];
  unsigned jw8[8];
#pragma unroll
  for (int q = 0; q < 8; ++q) {
    xx8[q] = __shfl(xxlane, q + 8 * hf, 32);           // |x|^2 of row (q + 8*hf)
    jw8[q] = jwbits[b * NN + n0 + q + 8 * hf];         // joint bitmask of that row
  }

  float rowacc[8];
#pragma unroll
  for (int q = 0; q < 8; ++q) rowacc[q] = __builtin_inff();

#pragma unroll 2
  for (int mt = wid; mt < MM / 16; mt += 4) {
    const int m0 = mt * 16;
    const float* tp = target_pc + ((size_t)b * MM + m0 + lcol) * 3;
    float tx = tp[0], ty = tp[1], tz = tp[2];
    float yy = tx * tx + ty * ty + tz * tz;            // |y_col|^2 for this lane's column
    v2f bf;
    bf.x = hf ? tz : tx;   // B 4x16: lanes 0-15 rows K=0,1; lanes 16-31 rows K=2,3
    bf.y = hf ? 0.f : ty;
    const int slab = seg[b * MM + m0 + lcol];          // column's joint label
    const unsigned selbit = 1u << slab;
    unsigned* jslot = &jd1[wid][slab * 16 + 8 * hf];

    v8f c = {};
    c = __builtin_amdgcn_wmma_f32_16x16x4_f32(false, a, false, bf, (short)0, c,
                                              false, false);

    float colmin = __builtin_inff();
    float jcol = 100.0f;
#pragma unroll
    for (int q = 0; q < 8; ++q) {
      float dd = xx8[q] + yy - 2.0f * c[q];            // squared distance (may be ~-0)
      float dm = (jw8[q] & selbit) ? dd : 100.0f;      // branchless masked value
      atomicMin(&jslot[q], __float_as_uint(dm));       // sentinel push is a no-op
      rowacc[q] = fminf(rowacc[q], dd);
      colmin = fminf(colmin, dd);
      jcol = fminf(jcol, dm);
    }
    // fold the two row-halves on the VALU (v_permlanex16), then push cross-block
    colmin = fminf(colmin, xor16f(colmin));
    jcol = fminf(jcol, xor16f(jcol));
    if (hf == 0) {
      atomicMin(&dist2b[b * MM + m0 + lcol], __float_as_uint(fmaxf(colmin, 0.f)));
      atomicMin(&jd2b[b * MM + m0 + lcol], __float_as_uint(fmaxf(jcol, 0.f)));
    }
  }

  // row mins: butterfly within each 16-lane half, fold waves through LDS
#pragma unroll
  for (int q = 0; q < 8; ++q) {
    float v = rowacc[q];
    v = fminf(v, __shfl_xor(v, 8, 16));
    v = fminf(v, __shfl_xor(v, 4, 16));
    v = fminf(v, __shfl_xor(v, 2, 16));
    v = fminf(v, __shfl_xor(v, 1, 16));
    rowacc[q] = fmaxf(v, 0.f);
  }
  if (lcol == 0) {
#pragma unroll
    for (int q = 0; q < 8; ++q)
      atomicMin(&rowminb[q + 8 * hf], __float_as_uint(rowacc[q]));
  }
  __syncthreads();

  // merge per-wave jd1 regions; sentinel (bit-exact 100.0) -> 0; sqrt after reduce
  for (int s = threadIdx.x; s < JJ * 16; s += blockDim.x) {
    unsigned mb = jd1[0][s];
    mb = min(mb, jd1[1][s]);
    mb = min(mb, jd1[2][s]);
    mb = min(mb, jd1[3][s]);
    if (mb != BITS_100) {
      float v = raw_sqrtf(__uint_as_float(mb));
      if (v != 0.f) atomicAdd(&jsumLds[s >> 4], v);
    }
  }
  __syncthreads();
  if (threadIdx.x < JJ) {
    float v = jsumLds[threadIdx.x];
    if (v != 0.f) atomicAdd(&jsum1[threadIdx.x], v);
  }
  if (threadIdx.x == 0) {
    float s = 0.f;
#pragma unroll
    for (int r = 0; r < 16; ++r) s += raw_sqrtf(__uint_as_float(rowminb[r]));
    atomicAdd(s1, s);
  }
}

// ---------------- final single-block reduction + small losses ----------------
__global__ __launch_bounds__(256) void finalize_kernel(
    const unsigned* __restrict__ dist2b, const unsigned* __restrict__ jd2b,
    const int* __restrict__ seg, const float* __restrict__ jsum1,
    const float* __restrict__ s1, const float* __restrict__ coords_pred,
    const float* __restrict__ target_coords, const float* __restrict__ quats,
    float* __restrict__ out) {
  __shared__ float jsum[JJ];
  __shared__ float red[256];
  __shared__ float res[3];  // sum(dist2), coord SSE, physical sum
  const int tid = threadIdx.x;
  if (tid < JJ) jsum[tid] = 0.f;
  __syncthreads();

  // --- dist2 sum + jd2 scatter-by-label (values are squared; sqrt here) ---
  float s2 = 0.f;
  for (int i = tid; i < BB * MM; i += 256) {
    s2 += raw_sqrtf(__uint_as_float(dist2b[i]));
    unsigned jb = jd2b[i];
    if (jb != BITS_100) {
      float v = raw_sqrtf(__uint_as_float(jb));
      if (v != 0.f) atomicAdd(&jsum[seg[i]], v);
    }
  }
  red[tid] = s2;
  __syncthreads();
  for (int off = 128; off > 0; off >>= 1) {
    if (tid < off) red[tid] += red[tid + off];
    __syncthreads();
  }
  if (tid == 0) res[0] = red[0];
  __syncthreads();

  // --- coordinate MSE (sum of squares) ---
  float cs = 0.f;
  for (int i = tid; i < BB * JJ * 3; i += 256) {
    float d = coords_pred[i] - target_coords[i];
    cs += d * d;
  }
  red[tid] = cs;
  __syncthreads();
  for (int off = 128; off > 0; off >>= 1) {
    if (tid < off) red[tid] += red[tid + off];
    __syncthreads();
  }
  if (tid == 0) res[1] = red[0];
  __syncthreads();

  // --- physical (Euler range) loss: sum over (b,j,axis) of relu terms ---
  float ps = 0.f;
  for (int i = tid; i < BB * JJ; i += 256) {
    const int j = i % JJ;
    const float* q = quats + (size_t)i * 4;
    float w = q[0], x = q[1], y = q[2], z = q[3];
    float inv = rsqrtf(w * w + x * x + y * y + z * z);
    w *= inv; x *= inv; y *= inv; z *= inv;
    float ang0 = atan2f(2.0f * (w * x + y * z), 1.0f - 2.0f * (x * x + y * y));
    float t = 2.0f * (w * y - z * x);
    t = fminf(fmaxf(t, -1.0f), 1.0f);
    float ang1 = asinf(t);
    float ang2 = atan2f(2.0f * (w * z + x * y), 1.0f - 2.0f * (y * y + z * z));
    float ang[3] = {ang0, ang1, ang2};
#pragma unroll
    for (int a = 0; a < 3; ++a) {
      ps += fmaxf(c_lo[j * 3 + a] - ang[a], 0.f);
      ps += fmaxf(ang[a] - c_hi[j * 3 + a], 0.f);
    }
  }
  red[tid] = ps;
  __syncthreads();
  for (int off = 128; off > 0; off >>= 1) {
    if (tid < off) red[tid] += red[tid + off];
    __syncthreads();
  }
  if (tid == 0) res[2] = red[0];
  __syncthreads();

  if (tid == 0) {
    const float invBN = 1.0f / (float)(BB * NN);
    float chamfer = (*s1 + res[0]) * invBN;
    float coord_loss = res[1] / (float)(BB * JJ * 3);
    float p_loss = res[2] / (float)(BB * JJ);
    float jmean = 0.f;
    for (int j = 0; j < JJ; ++j) {
      float jl = (jsum1[j] + jsum[j]) * invBN;
      out[2 + j] = jl;
      jmean += jl;
    }
    jmean /= (float)JJ;
    out[1] = coord_loss;
    out[23] = chamfer;
    out[24] = p_loss;
    out[0] = 0.1f * chamfer + 0.1f * jmean + coord_loss + 0.1f * p_loss;
  }
}

extern "C" void kernel_launch(void* const* d_in, const int* in_sizes, int n_in,
                              void* d_out, int out_size, void* d_ws, size_t ws_size,
                              hipStream_t stream) {
  const float* gen_pc = (const float*)d_in[0];
  const int* out_idxs = (const int*)d_in[1];
  const float* coords_pred = (const float*)d_in[2];
  const float* quats = (const float*)d_in[3];
  const float* seg_logits = (const float*)d_in[4];
  const float* bone_weights = (const float*)d_in[5];
  const float* target_coords = (const float*)d_in[6];
  const float* target_pc = (const float*)d_in[7];
  float* out = (float*)d_out;

  char* ws = (char*)d_ws;
  unsigned* dist2b = (unsigned*)ws;           ws += (size_t)BB * MM * 4;
  unsigned* jd2b = (unsigned*)ws;             ws += (size_t)BB * MM * 4;
  int* seg = (int*)ws;                        ws += (size_t)BB * MM * 4;
  unsigned* jwbits = (unsigned*)ws;           ws += (size_t)BB * NN * 4;
  float* jsum1 = (float*)ws;                  ws += JJ * 4;
  float* s1 = (float*)ws;                     ws += 4;

  const int tB = 256;
  init_kernel<<<(BB * MM + tB - 1) / tB, tB, 0, stream>>>(dist2b, jd2b, jsum1, s1);
  seg_kernel<<<(BB * MM + tB - 1) / tB, tB, 0, stream>>>(seg_logits, seg);
  jw_kernel<<<(BB * NN + tB - 1) / tB, tB, 0, stream>>>(out_idxs, bone_weights, jwbits);
  chamfer_kernel<<<BB * (NN / 16), 128, 0, stream>>>(
      gen_pc, target_pc, jwbits, seg, dist2b, jd2b, jsum1, s1);
  finalize_kernel<<<1, 256, 0, stream>>>(dist2b, jd2b, seg, jsum1, s1, coords_pred,
                                         target_coords, quats, out);
}